// SNN_59811714564168
// MI455X (gfx1250) — compile-verified
//
#include <hip/hip_runtime.h>
#include <cmath>

#define NI 784
#define NH 512
#define NO 10
#define BATCH 256
#define TSTEPS 64

typedef float v2f __attribute__((ext_vector_type(2)));
typedef float v8f __attribute__((ext_vector_type(8)));

#define V_L     (-70.0f)
#define V_RESET (-70.0f)
#define V_TH    (-50.0f)
#define G_L     (0.025f)
#define C_M     (0.5f)
#define INJ     (10.0f)

// ---------------- init: state, lag-weight table -------------------------
__global__ void k_init(float* __restrict__ h_mem, float* __restrict__ o_mem,
                       float* __restrict__ h_spk_prev, float* __restrict__ o_spk_prev,
                       float* __restrict__ wlag) {
    int i = blockIdx.x * blockDim.x + threadIdx.x;
    if (i < BATCH * NH) { h_mem[i] = V_L; h_spk_prev[i] = 0.f; }
    if (i < BATCH * NO) { o_mem[i] = V_L; o_spk_prev[i] = 0.f; }
    if (i < TSTEPS) {
        float lag = (float)i;
        wlag[i] = (i >= 1) ? (powf(lag + 1.f, 0.8f) - powf(lag, 0.8f)) : 0.f;
    }
}

// ------- repack W [nOut][K] into K-pair-interleaved B operand layout ----
// dst[((k/2)*nOut + j)*2 + (k&1)] = W[j][k]
// so a WMMA B operand {B[k][j], B[k+1][j]} is one contiguous b64 load.
__global__ void k_pack_pairs(const float* __restrict__ src, float* __restrict__ dst,
                             int nOut, int K) {
    int i = blockIdx.x * blockDim.x + threadIdx.x;
    if (i < nOut * K) {
        int j = i / K, k = i % K;
        dst[((size_t)(k >> 1) * nOut + j) * 2 + (k & 1)] = src[i];
    }
}

// ---------------- big fp32 WMMA GEMM: Icur = X @ W_fh^T + b_fh ----------
// M = T*B = 16384, N = NH = 512, K = NI = 784.  Wave -> 32x64 strip
// (2 M-tiles x 4 N-tiles = 8 WMMAs per K-step vs 6 b64 loads).
__global__ void __launch_bounds__(256) k_gemm_input(
        const float* __restrict__ data,     // [B][T][NI]
        const float* __restrict__ WfhPair,  // pair-interleaved [NI/2][NH][2]
        const float* __restrict__ b_fh,     // [NH]
        float* __restrict__ Icur)           // [T*B][NH]
{
    const int lane = threadIdx.x & 31;
    const int wave = threadIdx.x >> 5;
    const int tile = blockIdx.x * 8 + wave;   // 4096 waves total
    const int mGrp = tile >> 3;               // 0..511  (32 rows each)
    const int nGrp = tile & 7;                // 0..7    (64 cols each)
    const int m0   = mGrp * 32;
    const int t    = m0 / BATCH;
    const int b0   = m0 % BATCH;
    const int col0 = nGrp * 64;

    const int half = lane >> 4;   // selects K pair {k,k+1} vs {k+2,k+3}
    const int lid  = lane & 15;

    const float* Arow0 = data + (size_t)(b0 + lid)      * (TSTEPS * NI) + (size_t)t * NI;
    const float* Arow1 = data + (size_t)(b0 + 16 + lid) * (TSTEPS * NI) + (size_t)t * NI;
    __builtin_prefetch(Arow0, 0, 1);
    __builtin_prefetch(Arow1, 0, 1);

    v8f acc00 = {}, acc01 = {}, acc02 = {}, acc03 = {};
    v8f acc10 = {}, acc11 = {}, acc12 = {}, acc13 = {};
    for (int k = 0; k < NI; k += 4) {
        const int kb = k + half * 2;
        const int p  = kb >> 1;
        v2f a0 = *(const v2f*)(Arow0 + kb);
        v2f a1 = *(const v2f*)(Arow1 + kb);
        const float* Bp = WfhPair + ((size_t)p * NH + col0 + lid) * 2;
        v2f bv0 = *(const v2f*)(Bp);        // cols col0+lid
        v2f bv1 = *(const v2f*)(Bp + 32);   // +16 cols
        v2f bv2 = *(const v2f*)(Bp + 64);   // +32 cols
        v2f bv3 = *(const v2f*)(Bp + 96);   // +48 cols
        acc00 = __builtin_amdgcn_wmma_f32_16x16x4_f32(false, a0, false, bv0, (short)0, acc00, false, false);
        acc01 = __builtin_amdgcn_wmma_f32_16x16x4_f32(false, a0, false, bv1, (short)0, acc01, false, false);
        acc02 = __builtin_amdgcn_wmma_f32_16x16x4_f32(false, a0, false, bv2, (short)0, acc02, false, false);
        acc03 = __builtin_amdgcn_wmma_f32_16x16x4_f32(false, a0, false, bv3, (short)0, acc03, false, false);
        acc10 = __builtin_amdgcn_wmma_f32_16x16x4_f32(false, a1, false, bv0, (short)0, acc10, false, false);
        acc11 = __builtin_amdgcn_wmma_f32_16x16x4_f32(false, a1, false, bv1, (short)0, acc11, false, false);
        acc12 = __builtin_amdgcn_wmma_f32_16x16x4_f32(false, a1, false, bv2, (short)0, acc12, false, false);
        acc13 = __builtin_amdgcn_wmma_f32_16x16x4_f32(false, a1, false, bv3, (short)0, acc13, false, false);
    }
    // D layout: VGPR r, lanes 0-15 -> M=r, lanes 16-31 -> M=8+r; N = lid
    const float bb0 = b_fh[col0 +  0 + lid];
    const float bb1 = b_fh[col0 + 16 + lid];
    const float bb2 = b_fh[col0 + 32 + lid];
    const float bb3 = b_fh[col0 + 48 + lid];
    const int mrow0 = m0 + half * 8;
    const int mrow1 = m0 + 16 + half * 8;
    #pragma unroll
    for (int r = 0; r < 8; ++r) {
        size_t base = (size_t)(mrow0 + r) * NH + col0 + lid;
        Icur[base +  0] = acc00[r] + bb0;
        Icur[base + 16] = acc01[r] + bb1;
        Icur[base + 32] = acc02[r] + bb2;
        Icur[base + 48] = acc03[r] + bb3;
        base = (size_t)(mrow1 + r) * NH + col0 + lid;
        Icur[base +  0] = acc10[r] + bb0;
        Icur[base + 16] = acc11[r] + bb1;
        Icur[base + 32] = acc12[r] + bb2;
        Icur[base + 48] = acc13[r] + bb3;
    }
}

// ---------------- per-step hidden layer: WMMA recurrent GEMM + fLIF -----
__global__ void __launch_bounds__(256) k_hidden_step(
        const float* __restrict__ Icur,      // [T*B][NH] input current (+b_fh)
        const float* __restrict__ WlhPair,   // pair-interleaved [NH/2][NH][2]
        const float* __restrict__ W_fb,      // [NH][NO]
        const float* __restrict__ b_fb,      // [NH]
        const float* __restrict__ b_lh,      // [NH]
        const float* __restrict__ hs_prev,   // [B][NH]
        const float* __restrict__ os_prev,   // [B][NO]
        float* __restrict__ hs_cur,          // [B][NH]
        float* __restrict__ h_mem,           // [B][NH]
        float* __restrict__ h_dV,            // [T][B][NH]
        const float* __restrict__ wlag,      // [T]
        float frac_scale, int t)
{
    __shared__ float s_wlag[TSTEPS];
    if (threadIdx.x < TSTEPS) s_wlag[threadIdx.x] = wlag[threadIdx.x];
    __syncthreads();

    const int lane = threadIdx.x & 31;
    const int wave = threadIdx.x >> 5;
    const int tile = blockIdx.x * 8 + wave;   // 512 tiles of 16x16
    const int b0 = (tile >> 5) * 16;
    const int j0 = (tile & 31) * 16;
    const int half = lane >> 4;
    const int lid  = lane & 15;

    // recurrent term: hs_prev @ W_lh^T  (K = NH)
    v8f acc = {};
    {
        const float* Arow = hs_prev + (size_t)(b0 + lid) * NH;
        for (int k = 0; k < NH; k += 4) {
            const int kb = k + half * 2;
            v2f a = *(const v2f*)(Arow + kb);
            v2f b = *(const v2f*)(WlhPair + ((size_t)(kb >> 1) * NH + j0 + lid) * 2);
            acc = __builtin_amdgcn_wmma_f32_16x16x4_f32(false, a, false, b, (short)0, acc, false, false);
        }
    }

    const int j = j0 + lid;
    float wfb[NO];
    #pragma unroll
    for (int k = 0; k < NO; ++k) wfb[k] = W_fb[j * NO + k];
    const float bias = b_fb[j] + b_lh[j];

    const int bbase = b0 + half * 8;
    #pragma unroll
    for (int r = 0; r < 8; ++r) {
        const int bb = bbase + r;
        const size_t idx = (size_t)bb * NH + j;
        float cur = acc[r] + Icur[(size_t)(t * BATCH + bb) * NH + j] + bias;
        const float* osp = os_prev + bb * NO;
        #pragma unroll
        for (int k = 0; k < NO; ++k) cur += osp[k] * wfb[k];

        const float V = h_mem[idx];
        const float markov = frac_scale * (-G_L * (V - V_L) + cur) / C_M;
        float memory = 0.f;
        for (int lag = 1; lag <= t - 1; ++lag)
            memory += s_wlag[lag] * h_dV[(size_t)(t - lag) * (BATCH * NH) + idx];
        const float V_new = V + markov - memory;
        const bool fired = V_new > V_TH;
        const float V_post = fired ? V_RESET : V_new;
        hs_cur[idx] = fired ? 1.f : 0.f;
        h_mem[idx]  = V_post;
        h_dV[(size_t)t * (BATCH * NH) + idx] = V_post - V;
    }
}

// ---------------- per-step output layer (tiny: B*NO = 2560) -------------
__global__ void k_output_step(
        const float* __restrict__ hs_cur,   // [B][NH]
        const float* __restrict__ os_prev,  // [B][NO]
        const float* __restrict__ W_fo,     // [NO][NH]
        const float* __restrict__ b_fo,
        const float* __restrict__ W_lo,     // [NO][NO]
        const float* __restrict__ b_lo,
        const int*  __restrict__ targets,   // [B]
        float* __restrict__ os_cur,
        float* __restrict__ o_mem,
        float* __restrict__ o_dV,           // [T][B][NO]
        const float* __restrict__ wlag,
        float* __restrict__ out_spk,        // [T][B][NO]
        float* __restrict__ out_mem,        // [T][B][NO]
        float frac_scale, int t)
{
    const int i = blockIdx.x * blockDim.x + threadIdx.x;
    if (i >= BATCH * NO) return;
    const int b = i / NO, k = i % NO;

    const float4* hs = (const float4*)(hs_cur + (size_t)b * NH);
    const float4* wr = (const float4*)(W_fo + (size_t)k * NH);
    float cur = 0.f;
    for (int jj = 0; jj < NH / 4; ++jj) {
        float4 h = hs[jj], w = wr[jj];
        cur += h.x * w.x + h.y * w.y + h.z * w.z + h.w * w.w;
    }
    cur += b_fo[k] + b_lo[k];
    const float* osp = os_prev + b * NO;
    #pragma unroll
    for (int kk = 0; kk < NO; ++kk) cur += osp[kk] * W_lo[k * NO + kk];
    if (targets[b] == k) cur += INJ;

    const float V = o_mem[i];
    const float markov = frac_scale * (-G_L * (V - V_L) + cur) / C_M;
    float memory = 0.f;
    for (int lag = 1; lag <= t - 1; ++lag)
        memory += wlag[lag] * o_dV[(size_t)(t - lag) * (BATCH * NO) + i];
    const float V_new = V + markov - memory;
    const bool fired = V_new > V_TH;
    const float V_post = fired ? V_RESET : V_new;
    const float spk = fired ? 1.f : 0.f;

    os_cur[i] = spk;
    o_mem[i]  = V_post;
    o_dV[(size_t)t * (BATCH * NO) + i] = V_post - V;
    out_spk[(size_t)t * (BATCH * NO) + i] = spk;
    out_mem[(size_t)t * (BATCH * NO) + i] = V_post;
}

extern "C" void kernel_launch(void* const* d_in, const int* in_sizes, int n_in,
                              void* d_out, int out_size, void* d_ws, size_t ws_size,
                              hipStream_t stream) {
    const float* data    = (const float*)d_in[0];
    const int*   targets = (const int*)  d_in[1];
    // d_in[2] = plot (unused)
    const float* W_fh = (const float*)d_in[3];
    const float* b_fh = (const float*)d_in[4];
    const float* W_fo = (const float*)d_in[5];
    const float* b_fo = (const float*)d_in[6];
    const float* W_fb = (const float*)d_in[7];
    const float* b_fb = (const float*)d_in[8];
    const float* W_lh = (const float*)d_in[9];
    const float* b_lh = (const float*)d_in[10];
    const float* W_lo = (const float*)d_in[11];
    const float* b_lo = (const float*)d_in[12];

    float* ws = (float*)d_ws;
    size_t off = 0;
    float* Icur    = ws + off; off += (size_t)TSTEPS * BATCH * NH;   // 32 MB
    float* WfhPair = ws + off; off += (size_t)NI * NH;
    float* WlhPair = ws + off; off += (size_t)NH * NH;
    float* h_spk   = ws + off; off += 2 * (size_t)BATCH * NH;        // double buffer
    float* o_spk   = ws + off; off += 2 * (size_t)BATCH * NO;
    float* h_mem   = ws + off; off += (size_t)BATCH * NH;
    float* o_mem   = ws + off; off += (size_t)BATCH * NO;
    float* h_dV    = ws + off; off += (size_t)TSTEPS * BATCH * NH;   // 32 MB
    float* o_dV    = ws + off; off += (size_t)TSTEPS * BATCH * NO;
    float* wlag    = ws + off; off += TSTEPS;

    float* out_spk = (float*)d_out;
    float* out_mem = out_spk + (size_t)TSTEPS * BATCH * NO;

    // DT^alpha * Gamma(2-alpha), computed on host (deterministic)
    const float frac_scale = (float)(pow(0.1, 0.2) * tgamma(1.8));

    // one-time prep (parity buffer 1 = "previous spikes at t=0" = zeros)
    k_init<<<(BATCH * NH + 255) / 256, 256, 0, stream>>>(
        h_mem, o_mem, h_spk + (size_t)BATCH * NH, o_spk + (size_t)BATCH * NO, wlag);
    k_pack_pairs<<<(NH * NI + 255) / 256, 256, 0, stream>>>(W_fh, WfhPair, NH, NI);
    k_pack_pairs<<<(NH * NH + 255) / 256, 256, 0, stream>>>(W_lh, WlhPair, NH, NH);

    // hoisted input projection for all timesteps (full-occupancy fp32 WMMA GEMM)
    k_gemm_input<<<512, 256, 0, stream>>>(data, WfhPair, b_fh, Icur);

    // sequential fractional-LIF scan: 2 kernels per step give the grid-wide sync
    for (int t = 0; t < TSTEPS; ++t) {
        float*       hs_cur  = h_spk + (size_t)(t & 1) * (BATCH * NH);
        const float* hs_prev = h_spk + (size_t)((t + 1) & 1) * (BATCH * NH);
        float*       os_cur  = o_spk + (size_t)(t & 1) * (BATCH * NO);
        const float* os_prev = o_spk + (size_t)((t + 1) & 1) * (BATCH * NO);

        k_hidden_step<<<64, 256, 0, stream>>>(
            Icur, WlhPair, W_fb, b_fb, b_lh, hs_prev, os_prev,
            hs_cur, h_mem, h_dV, wlag, frac_scale, t);

        k_output_step<<<(BATCH * NO + 255) / 256, 256, 0, stream>>>(
            hs_cur, os_prev, W_fo, b_fo, W_lo, b_lo, targets,
            os_cur, o_mem, o_dV, wlag, out_spk, out_mem, frac_scale, t);
    }
}